// ConvSingleNodeModel_12000138625377
// MI455X (gfx1250) — compile-verified
//
#include <hip/hip_runtime.h>
#include <hip/hip_bf16.h>

typedef __attribute__((ext_vector_type(2))) float v2f;
typedef __attribute__((ext_vector_type(8))) float v8f;

#define NNODES 50000
#define NEDGES 800000
#define KDIM   128
#define CH     128
#define NCOUT  64
#define BNEPS  1e-5f

// ---------------- degree / normalization ----------------

__global__ void deg_init_kernel(float* __restrict__ deg, int n) {
    int i = blockIdx.x * blockDim.x + threadIdx.x;
    if (i < n) deg[i] = 1.0f;   // self-loop weight
}

__global__ void deg_edge_kernel(const int* __restrict__ ei, const float* __restrict__ ew,
                                float* __restrict__ deg, int e) {
    int i = blockIdx.x * blockDim.x + threadIdx.x;
    if (i < e) unsafeAtomicAdd(&deg[ei[e + i]], ew[i]);   // dst = ei[E + i]
}

__global__ void dis_kernel(const float* __restrict__ deg, float* __restrict__ dis, int n) {
    int i = blockIdx.x * blockDim.x + threadIdx.x;
    if (i < n) {
        float d = deg[i];
        dis[i] = d > 0.0f ? rsqrtf(d) : 0.0f;
    }
}

__global__ void norm_kernel(const int* __restrict__ ei, const float* __restrict__ ew,
                            const float* __restrict__ dis, float* __restrict__ nrm, int e) {
    int i = blockIdx.x * blockDim.x + threadIdx.x;
    if (i < e) nrm[i] = dis[ei[i]] * ew[i] * dis[ei[e + i]];
}

// ---- WMMA GEMM: out[n,CO] = A[n,128] @ W[128,CO] (+bias)(+lrelu) ----
// W staged in LDS k-interleaved: sW[((k>>2)*CO + col)*4 + (k&3)] = W[k][col]
// so each lane's B operand (W[k+kh][col], W[k+kh+1][col]) is one contiguous b64,
// and the 32 lanes of a wave hit all 64 LDS banks exactly once (conflict-free).

template <int CO, bool BIAS, bool LRELU>
__global__ __launch_bounds__(256) void gemm_wmma_kernel(const float* __restrict__ A,
                                                        const float* __restrict__ W,
                                                        const float* __restrict__ bias,
                                                        float* __restrict__ out, int nrows) {
    constexpr int T = CO / 16;
    __shared__ float sW[KDIM * CO];   // 64 KB (CO=128) / 32 KB (CO=64); WGP has 320 KB

    // cooperative interleaved stage of W (all threads reach the barrier before any exit)
    for (int i4 = threadIdx.x; i4 < (KDIM * CO) / 4; i4 += 256) {
        const int idx = i4 * 4;
        const int kk  = idx / CO;         // constexpr CO -> shifts
        const int c   = idx % CO;         // 4 consecutive cols, same k row
        const float4 wv = *reinterpret_cast<const float4*>(W + idx);
        float* dp = sW + (((kk >> 2) * CO + c) << 2) + (kk & 3);
        dp[0]  = wv.x;
        dp[4]  = wv.y;
        dp[8]  = wv.z;
        dp[12] = wv.w;
    }
    __syncthreads();

    const int lane  = threadIdx.x & 31;
    const int wave  = threadIdx.x >> 5;
    const int strip = blockIdx.x * 8 + wave;
    const int rb    = strip * 16;
    if (rb >= nrows) return;

    const int l  = lane & 15;          // row (A) / col (B) within tile
    const int kh = (lane >> 4) << 1;   // 0 for lanes 0-15, 2 for lanes 16-31

    v8f acc[T];
#pragma unroll
    for (int t = 0; t < T; ++t) acc[t] = 0.0f;

    const float* arow = A + (size_t)(rb + l) * KDIM + kh;

    for (int k = 0; k < KDIM; k += 4) {
        float2 av = *reinterpret_cast<const float2*>(arow + k);
        v2f a;
        a.x = av.x;   // K = k + kh
        a.y = av.y;   // K = k + kh + 1
        const float* bbase = sW + k * CO + kh;   // == ((k>>2)*CO)<<2, + kh
#pragma unroll
        for (int t = 0; t < T; ++t) {
            v2f b = *reinterpret_cast<const v2f*>(bbase + ((t * 16 + l) << 2));
            acc[t] = __builtin_amdgcn_wmma_f32_16x16x4_f32(false, a, false, b,
                                                           (short)0, acc[t], false, false);
        }
    }

    const int row_off = (lane >> 4) * 8;   // C/D layout: lanes 16-31 hold M = r + 8
#pragma unroll
    for (int t = 0; t < T; ++t) {
        const int col = t * 16 + l;
        float bv = BIAS ? bias[col] : 0.0f;
#pragma unroll
        for (int r = 0; r < 8; ++r) {
            int row = rb + r + row_off;
            float v = acc[t][r] + bv;
            if (LRELU) v = v > 0.0f ? v : 0.01f * v;
            out[(size_t)row * CO + col] = v;
        }
    }
}

// ---------------- GCN aggregate: init (self loop + bias) then edge scatter ----------------

__global__ void agg_init_kernel(const float* __restrict__ Hm, const float* __restrict__ dis,
                                const float* __restrict__ bias, float* __restrict__ agg, int total) {
    int i = blockIdx.x * blockDim.x + threadIdx.x;
    if (i >= total) return;
    int node = i >> 7;
    int c    = i & 127;
    float d  = dis[node];
    agg[i] = d * d * Hm[i] + bias[c];
}

__global__ __launch_bounds__(256) void scatter_kernel(const int* __restrict__ ei,
                                                      const float* __restrict__ nrm,
                                                      const float* __restrict__ Hm,
                                                      float* __restrict__ agg, int e) {
    const int lane = threadIdx.x & 31;
    const int edge = blockIdx.x * 8 + (threadIdx.x >> 5);
    if (edge >= e) return;
    const int src = ei[edge];
    const int dst = ei[e + edge];
    const float w = nrm[edge];
    const float4 hv = *reinterpret_cast<const float4*>(Hm + (size_t)src * CH + lane * 4);
    float* d = agg + (size_t)dst * CH + lane * 4;
    unsafeAtomicAdd(d + 0, w * hv.x);
    unsafeAtomicAdd(d + 1, w * hv.y);
    unsafeAtomicAdd(d + 2, w * hv.z);
    unsafeAtomicAdd(d + 3, w * hv.w);
}

// ---------------- BN (eval) + optional residual + leaky-relu ----------------

__global__ void bn_act_kernel(const float* __restrict__ agg, const float* __restrict__ g,
                              const float* __restrict__ bb, const float* __restrict__ m,
                              const float* __restrict__ v, const float* __restrict__ resid,
                              float* __restrict__ out, int total) {
    int i = blockIdx.x * blockDim.x + threadIdx.x;
    if (i >= total) return;
    int c = i & 127;
    float s   = g[c] * rsqrtf(v[c] + BNEPS);
    float val = (agg[i] - m[c]) * s + bb[c];
    if (resid) val += resid[i];
    out[i] = val > 0.0f ? val : 0.01f * val;
}

// ---------------- launcher ----------------

extern "C" void kernel_launch(void* const* d_in, const int* in_sizes, int n_in,
                              void* d_out, int out_size, void* d_ws, size_t ws_size,
                              hipStream_t stream) {
    const float* x      = (const float*)d_in[0];
    const int*   ei     = (const int*)d_in[1];
    const float* ew     = (const float*)d_in[2];
    const float* w1     = (const float*)d_in[3];
    const float* b1     = (const float*)d_in[4];
    const float* bn1_g  = (const float*)d_in[5];
    const float* bn1_b  = (const float*)d_in[6];
    const float* bn1_m  = (const float*)d_in[7];
    const float* bn1_v  = (const float*)d_in[8];
    const float* conv_ws = (const float*)d_in[9];
    const float* conv_bs = (const float*)d_in[10];
    const float* bns_g  = (const float*)d_in[11];
    const float* bns_b  = (const float*)d_in[12];
    const float* bns_m  = (const float*)d_in[13];
    const float* bns_v  = (const float*)d_in[14];
    const float* lin1_w = (const float*)d_in[15];
    const float* lin1_b = (const float*)d_in[16];
    const float* lin2_w = (const float*)d_in[17];
    const float* lin2_b = (const float*)d_in[18];

    const int N = NNODES, E = NEDGES;
    const int total = N * CH;   // 6.4M, fits int

    float* ws  = (float*)d_ws;
    float* deg = ws;                          // N
    float* dis = deg + N;                     // N
    float* nrm = dis + N;                     // E
    float* B0  = nrm + E;                     // N*CH  (H = h @ W)
    float* B1  = B0 + (size_t)N * CH;         // N*CH  (aggregate)
    float* B2  = B1 + (size_t)N * CH;         // N*CH  (current features)
    float* out = (float*)d_out;

    const int TB = 256;
    const int gElem  = (total + TB - 1) / TB;
    const int gNode  = (N + TB - 1) / TB;
    const int gEdge  = (E + TB - 1) / TB;
    const int gScat  = (E + 7) / 8;           // 8 edges per 256-thread block
    const int gGemm  = (N / 16 + 7) / 8;      // 8 waves per block, 1 strip each

    // symmetric degree normalization
    deg_init_kernel<<<gNode, TB, 0, stream>>>(deg, N);
    deg_edge_kernel<<<gEdge, TB, 0, stream>>>(ei, ew, deg, E);
    dis_kernel<<<gNode, TB, 0, stream>>>(deg, dis, N);
    norm_kernel<<<gEdge, TB, 0, stream>>>(ei, ew, dis, nrm, E);

    // layer 1: h = lrelu(bn1(gcn(x, w1, b1)))
    gemm_wmma_kernel<CH, false, false><<<gGemm, TB, 0, stream>>>(x, w1, nullptr, B0, N);
    agg_init_kernel<<<gElem, TB, 0, stream>>>(B0, dis, b1, B1, total);
    scatter_kernel<<<gScat, TB, 0, stream>>>(ei, nrm, B0, B1, E);
    bn_act_kernel<<<gElem, TB, 0, stream>>>(B1, bn1_g, bn1_b, bn1_m, bn1_v, nullptr, B2, total);

    // residual GCN layers
    for (int i = 0; i < 2; ++i) {
        gemm_wmma_kernel<CH, false, false><<<gGemm, TB, 0, stream>>>(B2, conv_ws + (size_t)i * CH * CH,
                                                                     nullptr, B0, N);
        agg_init_kernel<<<gElem, TB, 0, stream>>>(B0, dis, conv_bs + i * CH, B1, total);
        scatter_kernel<<<gScat, TB, 0, stream>>>(ei, nrm, B0, B1, E);
        bn_act_kernel<<<gElem, TB, 0, stream>>>(B1, bns_g + i * CH, bns_b + i * CH,
                                                bns_m + i * CH, bns_v + i * CH, B2, B2, total);
    }

    // head: lrelu(h @ lin1_w + b) then h @ lin2_w + b
    gemm_wmma_kernel<CH, true, true><<<gGemm, TB, 0, stream>>>(B2, lin1_w, lin1_b, B0, N);
    gemm_wmma_kernel<NCOUT, true, false><<<gGemm, TB, 0, stream>>>(B0, lin2_w, lin2_b, out, N);
}